// HBVMulET_1_1p_57208964383106
// MI455X (gfx1250) — compile-verified
//
#include <hip/hip_runtime.h>
#include <hip/hip_bf16.h>

// HBV rainfall-runoff scan: 1000 sequential steps over 64000 (grid,mu) cells.
// One thread per cell; forcing data (P,T,ET) double-buffer prefetched into LDS
// with CDNA5 async global->LDS copies; mu-mean via wave32 shfl_xor reduction.
// Single-log trick: log(min(SM,FC)) = min(log SM, log FC) -> 1 log + 2 exp per
// step instead of 2 logs + 2 exps.

#define CH    64      // time steps per prefetch chunk
#define CELLS 16      // grid cells per block
#define MU    16
#define TPB   256
#define PRECS 1e-5f

__global__ __launch_bounds__(TPB)
void hbv_scan_kernel(const float* __restrict__ x,
                     const float* __restrict__ params,
                     float* __restrict__ out,
                     int Nstep, int Ngrid)
{
    // 2 x (64 steps x 16 cells x 3 floats) = 24 KB
    __shared__ float xbuf[2][CH * CELLS * 3];

    const int local  = threadIdx.x;
    const int m      = local & (MU - 1);
    const int gLocal = local >> 4;
    const int g      = blockIdx.x * CELLS + gLocal;

    // ---- load + scale 14 parameters into registers ----
    const float lo[14] = {1.f, 50.f, 0.05f, 0.01f, 0.001f, 0.2f, 0.f, 0.f,
                          -2.5f, 0.5f, 0.f, 0.f, 0.3f, 0.f};
    const float hi[14] = {6.f, 1000.f, 0.9f, 0.5f, 0.2f, 1.f, 10.f, 100.f,
                          2.5f, 10.f, 0.1f, 0.2f, 5.f, 1.f};
    float pr[14];
    const float* pbase = params + (size_t)g * 14 * MU + m;
#pragma unroll
    for (int i = 0; i < 14; ++i)
        pr[i] = lo[i] + pbase[i * MU] * (hi[i] - lo[i]);

    const float parBETA = pr[0],  parFC  = pr[1],  parK0  = pr[2];
    const float parK1   = pr[3],  parK2  = pr[4],  parLP  = pr[5];
    const float parPERC = pr[6],  parUZL = pr[7],  parTT  = pr[8];
    const float parCFMAX= pr[9],  parCFR = pr[10], parCWH = pr[11];
    const float parBETAET = pr[12], parC = pr[13];

    const float invFC     = 1.0f / parFC;
    const float kRefreeze = parCFR * parCFMAX;
    // hoisted transcendentals (per thread, once for all 1000 steps)
    const float logFC   = __logf(parFC);
    const float logLPFC = __logf(parLP * parFC);

    float SNOWPACK = 0.001f, MELTWATER = 0.001f, SM = 0.001f,
          SUZ = 0.001f, SLZ = 0.001f;

    // ---- async prefetch bookkeeping ----
    // generic shared pointer: low 32 bits == LDS byte offset
    const uint32_t ldsBase   = (uint32_t)(uintptr_t)(&xbuf[0][0]);
    const uint64_t gBase     = (uint64_t)(uintptr_t)x
                             + (uint64_t)blockIdx.x * (CELLS * 3 * 4); // 192 B
    const uint64_t stepBytes = (uint64_t)Ngrid * 12u;  // 3 floats * 4 B * Ngrid
    const int nchunks = (Nstep + CH - 1) / CH;

    // issue chunk 0: 768 B128 copies, exactly 3 per thread (uniform per wave)
#pragma unroll
    for (int r = 0; r < 3; ++r) {
        int j = local + r * TPB;          // 0..767
        int s = j / 12, w = j % 12;
        int t = s; if (t > Nstep - 1) t = Nstep - 1;   // clamp (padding slots)
        uint64_t ga = gBase + (uint64_t)t * stepBytes + (uint32_t)(w * 16);
        uint32_t la = ldsBase + (uint32_t)(s * 192 + w * 16);
        asm volatile("global_load_async_to_lds_b128 %0, %1, off"
                     :: "v"(la), "v"(ga) : "memory");
    }

    const float inv16 = 1.0f / 16.0f;

    for (int c = 0; c < nchunks; ++c) {
        const int t0  = c * CH;
        const int buf = c & 1;

        if (c + 1 < nchunks) {
            // issue next chunk into the other buffer (its readers finished at
            // the barrier closing iteration c-1)
            const int t0n = (c + 1) * CH;
            const uint32_t lb = ldsBase
                              + (uint32_t)(((c + 1) & 1) * (CH * CELLS * 3 * 4));
#pragma unroll
            for (int r = 0; r < 3; ++r) {
                int j = local + r * TPB;
                int s = j / 12, w = j % 12;
                int t = t0n + s; if (t > Nstep - 1) t = Nstep - 1;
                uint64_t ga = gBase + (uint64_t)t * stepBytes + (uint32_t)(w * 16);
                uint32_t la = lb + (uint32_t)(s * 192 + w * 16);
                asm volatile("global_load_async_to_lds_b128 %0, %1, off"
                             :: "v"(la), "v"(ga) : "memory");
            }
            // 3 newly issued may remain; the 3 for chunk c must be done
            // (ASYNCcnt completes in issue order for async loads)
            asm volatile("s_wait_asynccnt 3" ::: "memory");
        } else {
            asm volatile("s_wait_asynccnt 0" ::: "memory");
        }
        __syncthreads();   // all waves' chunk-c copies visible in LDS

        const int steps = min(CH, Nstep - t0);
        const float* xs = &xbuf[buf][gLocal * 3];

#pragma unroll 2
        for (int s = 0; s < steps; ++s) {
            const float P = xs[s * 48 + 0];
            const float T = xs[s * 48 + 1];
            const float E = xs[s * 48 + 2];

            // ---- snow routine ----
            const bool  rain = (T >= parTT);
            const float RAIN = rain ? P : 0.0f;
            const float SNOW = rain ? 0.0f : P;
            SNOWPACK += SNOW;
            float melt = fminf(fmaxf(parCFMAX * (T - parTT), 0.0f), SNOWPACK);
            MELTWATER += melt;
            SNOWPACK  -= melt;
            float refreeze = fminf(fmaxf(kRefreeze * (parTT - T), 0.0f), MELTWATER);
            SNOWPACK  += refreeze;
            MELTWATER -= refreeze;
            float tosoil = fmaxf(MELTWATER - parCWH * SNOWPACK, 0.0f);
            MELTWATER -= tosoil;

            // ---- soil routine (single log; exps off the log) ----
            // SM >= PRECS > 0 always -> log finite
            const float lsm = __logf(SM);
            float soil_wetness = fminf(__expf(parBETA * (lsm - logFC)), 1.0f);
            float recharge = (RAIN + tosoil) * soil_wetness;
            float excess = fmaxf(SM - parFC, 0.0f);
            SM -= excess;                          // SM = min(SM, FC)
            // log(min(SM,FC)) = min(log SM, log FC)  (log monotone)
            const float lsm2 = fminf(lsm, logFC);
            float evapfactor = fminf(__expf(parBETAET * (lsm2 - logLPFC)), 1.0f);
            float ETact = fminf(SM, E * evapfactor);
            SM = fmaxf(SM - ETact, PRECS);
            SM = SM + RAIN + tosoil - recharge;
            float capillary = fminf(SLZ, parC * SLZ * (1.0f - fminf(SM * invFC, 1.0f)));
            SM  = fmaxf(SM + capillary, PRECS);
            SLZ = fmaxf(SLZ - capillary, PRECS);

            // ---- response routine ----
            SUZ += recharge + excess;
            float PERC = fminf(SUZ, parPERC);
            SUZ -= PERC;
            float Q0 = parK0 * fmaxf(SUZ - parUZL, 0.0f);
            SUZ -= Q0;
            float Q1 = parK1 * SUZ;
            SUZ -= Q1;
            SLZ += PERC;
            float Q2 = parK2 * SLZ;
            SLZ -= Q2;

            // ---- mean over mu (16 lanes) via wave32 xor-shuffles ----
            float q = Q0 + Q1 + Q2;
            q += __shfl_xor(q, 8, 32);
            q += __shfl_xor(q, 4, 32);
            q += __shfl_xor(q, 2, 32);
            q += __shfl_xor(q, 1, 32);
            if (m == 0)
                __builtin_nontemporal_store(q * inv16,
                                            &out[(size_t)(t0 + s) * Ngrid + g]);
        }
        __syncthreads();   // buffer 'buf' safe to overwrite next iteration
    }
}

extern "C" void kernel_launch(void* const* d_in, const int* in_sizes, int n_in,
                              void* d_out, int out_size, void* d_ws, size_t ws_size,
                              hipStream_t stream) {
    const float* x      = (const float*)d_in[0];
    const float* params = (const float*)d_in[1];
    (void)n_in; (void)d_ws; (void)ws_size; (void)out_size;

    const int mu    = 16;                         // per setup_inputs
    const int Ngrid = in_sizes[1] / (14 * mu);    // 4000
    const int Nstep = in_sizes[0] / (3 * Ngrid);  // 1000
    float* out = (float*)d_out;

    dim3 grid(Ngrid / CELLS);   // 250 blocks
    dim3 block(TPB);            // 256 threads = 16 cells x 16 mu
    hbv_scan_kernel<<<grid, block, 0, stream>>>(x, params, out, Nstep, Ngrid);
}